// GLMEdgeConv_33998961115202
// MI455X (gfx1250) — compile-verified
//
#include <hip/hip_runtime.h>
#include <stdint.h>

typedef __attribute__((ext_vector_type(2))) float v2f;
typedef __attribute__((ext_vector_type(8))) float v8f;

#define BATCH 4
#define NPTS  8192
#define CIN   64
#define KS    6
#define KL    12
#define EPSV  1e-5f
#define OUTSZ (BATCH * 128 * NPTS)

// stats region: per stream = 4x[64 partials x 64ch] + 4x64 scale/shift
#define STREAM_STRIDE 16640
#define STATSZ (2 * STREAM_STRIDE)

// ---------------------------------------------------------------------------
// Zero output + stats accumulators (harness poisons buffers; we must init).
// ---------------------------------------------------------------------------
__global__ __launch_bounds__(256) void zero_kernel(float* __restrict__ out,
                                                   float* __restrict__ stats) {
  int i = blockIdx.x * 256 + threadIdx.x;
  if (i < OUTSZ) out[i] = 0.0f;
  if (i < STATSZ) stats[i] = 0.0f;
}

// ---------------------------------------------------------------------------
// kNN: one thread per query point; candidate tiles staged global->LDS with
// CDNA5 async-to-LDS copies (ASYNCcnt), top-13 kept in a register-resident
// fully unrolled insertion sort (shared by k=6 and k=12 streams).
// ---------------------------------------------------------------------------
__global__ __launch_bounds__(256) void knn_kernel(const float* __restrict__ pos,
                                                  int* __restrict__ idxS,
                                                  int* __restrict__ idxL) {
  __shared__ float cx[256], cy[256], cz[256];
  const int b = blockIdx.x >> 5;                     // N/256 = 32 blocks per batch
  const int q = ((blockIdx.x & 31) << 8) + threadIdx.x;
  const float* pb = pos + (size_t)b * 3 * NPTS;
  const float px = pb[q], py = pb[NPTS + q], pz = pb[2 * NPTS + q];

  const unsigned ldsX = (unsigned)(uintptr_t)(void*)&cx[threadIdx.x];
  const unsigned ldsY = (unsigned)(uintptr_t)(void*)&cy[threadIdx.x];
  const unsigned ldsZ = (unsigned)(uintptr_t)(void*)&cz[threadIdx.x];

  float dist[13];
  int ind[13];
#pragma unroll
  for (int i = 0; i < 13; ++i) { dist[i] = 3.0e38f; ind[i] = 0; }

  for (int ct = 0; ct < NPTS / 256; ++ct) {
    const int cand = ct * 256 + threadIdx.x;
    {
      const unsigned long long gaX = (unsigned long long)(uintptr_t)(pb + cand);
      const unsigned long long gaY = gaX + (unsigned long long)NPTS * 4ull;
      const unsigned long long gaZ = gaY + (unsigned long long)NPTS * 4ull;
      asm volatile("global_load_async_to_lds_b32 %0, %1, off"
                   :: "v"(ldsX), "v"(gaX) : "memory");
      asm volatile("global_load_async_to_lds_b32 %0, %1, off"
                   :: "v"(ldsY), "v"(gaY) : "memory");
      asm volatile("global_load_async_to_lds_b32 %0, %1, off"
                   :: "v"(ldsZ), "v"(gaZ) : "memory");
      asm volatile("s_wait_asynccnt 0x0" ::: "memory");
    }
    // prefetch next candidate tile while this one is processed
    if (ct + 1 < NPTS / 256) {
      __builtin_prefetch(pb + cand + 256, 0, 0);
      __builtin_prefetch(pb + NPTS + cand + 256, 0, 0);
      __builtin_prefetch(pb + 2 * NPTS + cand + 256, 0, 0);
    }
    __syncthreads();
    for (int j = 0; j < 256; ++j) {
      const float dx = px - cx[j], dy = py - cy[j], dz = pz - cz[j];
      const float d2 = dx * dx + dy * dy + dz * dz;
      if (d2 < dist[12]) {
        dist[12] = d2; ind[12] = ct * 256 + j;
#pragma unroll
        for (int t = 12; t > 0; --t) {
          if (dist[t] < dist[t - 1]) {
            float td = dist[t]; dist[t] = dist[t - 1]; dist[t - 1] = td;
            int   ti = ind[t];  ind[t]  = ind[t - 1];  ind[t - 1]  = ti;
          }
        }
      }
    }
    __syncthreads();
  }
  // entry 0 is self (d2 == 0) -> dropped, matching reference idx[..., 1:]
#pragma unroll
  for (int i = 0; i < KS; ++i) idxS[((size_t)b * NPTS + q) * KS + i] = ind[1 + i];
#pragma unroll
  for (int i = 0; i < KL; ++i) idxL[((size_t)b * NPTS + q) * KL + i] = ind[1 + i];
}

// ---------------------------------------------------------------------------
// Finalize BN from 64 partial bins: scale = g*rsqrt(var+eps), shift = t-m*scale
// ---------------------------------------------------------------------------
__global__ __launch_bounds__(64) void bn_finalize_kernel(
    const float* __restrict__ sumP, const float* __restrict__ sqP,
    const float* __restrict__ g, const float* __restrict__ t, float cnt,
    float* __restrict__ scale, float* __restrict__ shift) {
  const int c = threadIdx.x;
  float s = 0.0f, q = 0.0f;
  for (int p = 0; p < 64; ++p) { s += sumP[p * 64 + c]; q += sqP[p * 64 + c]; }
  const float m = s / cnt;
  float v = q / cnt - m * m;
  v = v < 0.0f ? 0.0f : v;
  const float sc = g[c] * rsqrtf(v + EPSV);
  scale[c] = sc;
  shift[c] = t[c] - m * sc;
}

// Cross-lane (16-wide) column reduction of the C/D tile; atomics spread over
// 64 partial bins (blockIdx % 64) to avoid hot-cacheline serialization in L2.
__device__ __forceinline__ void wave_stats_atomic(const v8f* acc, int half,
                                                  int sN, int slot,
                                                  float* sumP, float* sqP) {
#pragma unroll
  for (int mt = 0; mt < 4; ++mt) {
#pragma unroll
    for (int r = 0; r < 8; ++r) {
      float v = acc[mt][r];
      float v2 = v * v;
#pragma unroll
      for (int msk = 8; msk >= 1; msk >>= 1) {
        v  += __shfl_xor(v,  msk, 32);
        v2 += __shfl_xor(v2, msk, 32);
      }
      if (sN == 0) {
        const int ch = mt * 16 + r + half * 8;
        atomicAdd(&sumP[slot * 64 + ch], v);
        atomicAdd(&sqP[slot * 64 + ch], v2);
      }
    }
  }
}

// ---------------------------------------------------------------------------
// EdgeConv pass (4 waves/block, 16 samples per wave, full f32 WMMA 16x16x4).
// MODE 0: conv1 -> accumulate layer-1 stats.
// MODE 1: conv1 -> BN1/ReLU -> conv2 -> accumulate layer-2 stats.
// MODE 2: full chain -> BN2/ReLU -> segmented max over k + atomicMax.
// ---------------------------------------------------------------------------
template <int MODE>
__global__ __launch_bounds__(128) void edgeconv_pass(
    const float* __restrict__ x, const int* __restrict__ idx, int k,
    const float* __restrict__ W1, const float* __restrict__ b1,
    const float* __restrict__ scale1, const float* __restrict__ shift1,
    const float* __restrict__ W2, const float* __restrict__ b2,
    const float* __restrict__ scale2, const float* __restrict__ shift2,
    float* __restrict__ sumP, float* __restrict__ sqP,
    float* __restrict__ out, int chOff) {
  __shared__ float eT[4][128 * 16];  // 32 KB: per-wave 128ch x 16-sample tile
  const int w = threadIdx.x >> 5;
  const int lane = threadIdx.x & 31;
  const int sN = lane & 15;   // column / A-row within 16
  const int half = lane >> 4; // K half-select for A/B, M half-select for C/D
  const int nk = NPTS * k;
  const int waveBase = blockIdx.x * 64 + w * 16;

  // ---- stage edge features e = [x_i ; x_j - x_i] into LDS (coalesced) ----
  {
    const int sample = waveBase + sN;
    const int b = sample / nk;
    const int rem = sample - b * nk;
    const int n = rem / k;
    const int kk = rem - n * k;
    const int jn = idx[((size_t)b * NPTS + n) * k + kk];
    const float* xb = x + (size_t)b * CIN * NPTS;
    float* e = eT[w];
#pragma unroll 8
    for (int c = half * 32; c < half * 32 + 32; ++c) {
      const float xi = xb[c * NPTS + n];
      const float xj = xb[c * NPTS + jn];
      e[c * 16 + sN] = xi;
      e[(64 + c) * 16 + sN] = xj - xi;
    }
  }
  __syncthreads();

  const float* e = eT[w];
  const int koff = half * 2;
  const int slot = blockIdx.x & 63;

  // ---- conv1: 64x16 tile = W1(64x128) x e(128x16), 32 K-chunks ----
  v8f acc[4];
  {
    v8f z = {0.f, 0.f, 0.f, 0.f, 0.f, 0.f, 0.f, 0.f};
    acc[0] = z; acc[1] = z; acc[2] = z; acc[3] = z;
  }
  for (int kc = 0; kc < 32; ++kc) {
    const int c0 = kc * 4;
    v2f bf;
    bf.x = e[(c0 + koff) * 16 + sN];
    bf.y = e[(c0 + koff + 1) * 16 + sN];
#pragma unroll
    for (int mt = 0; mt < 4; ++mt) {
      const int m = mt * 16 + sN;
      v2f af;
      af.x = W1[m * 128 + c0 + koff];
      af.y = W1[m * 128 + c0 + koff + 1];
      acc[mt] = __builtin_amdgcn_wmma_f32_16x16x4_f32(
          false, af, false, bf, (short)0, acc[mt], false, false);
    }
  }
#pragma unroll
  for (int mt = 0; mt < 4; ++mt)
#pragma unroll
    for (int r = 0; r < 8; ++r) acc[mt][r] += b1[mt * 16 + r + half * 8];

  if (MODE == 0) {
    wave_stats_atomic(acc, half, sN, slot, sumP, sqP);
    return;
  }

  // ---- BN1 + ReLU -> a1 staged back to LDS (layout change C/D -> B) ----
  __syncthreads();
  {
    float* a = eT[w];
#pragma unroll
    for (int mt = 0; mt < 4; ++mt)
#pragma unroll
      for (int r = 0; r < 8; ++r) {
        const int ch = mt * 16 + r + half * 8;
        float v = acc[mt][r] * scale1[ch] + shift1[ch];
        a[ch * 16 + sN] = v > 0.f ? v : 0.f;
      }
  }
  __syncthreads();

  // ---- conv2: 64x16 tile = W2(64x64) x a1(64x16), 16 K-chunks ----
  v8f acc2[4];
  {
    v8f z = {0.f, 0.f, 0.f, 0.f, 0.f, 0.f, 0.f, 0.f};
    acc2[0] = z; acc2[1] = z; acc2[2] = z; acc2[3] = z;
  }
  for (int kc = 0; kc < 16; ++kc) {
    const int c0 = kc * 4;
    v2f bf;
    bf.x = e[(c0 + koff) * 16 + sN];
    bf.y = e[(c0 + koff + 1) * 16 + sN];
#pragma unroll
    for (int mt = 0; mt < 4; ++mt) {
      const int m = mt * 16 + sN;
      v2f af;
      af.x = W2[m * 64 + c0 + koff];
      af.y = W2[m * 64 + c0 + koff + 1];
      acc2[mt] = __builtin_amdgcn_wmma_f32_16x16x4_f32(
          false, af, false, bf, (short)0, acc2[mt], false, false);
    }
  }
#pragma unroll
  for (int mt = 0; mt < 4; ++mt)
#pragma unroll
    for (int r = 0; r < 8; ++r) acc2[mt][r] += b2[mt * 16 + r + half * 8];

  if (MODE == 1) {
    wave_stats_atomic(acc2, half, sN, slot, sumP, sqP);
    return;
  }

  // ---- MODE 2: BN2 + ReLU; segmented (equal-n run) suffix-max across the
  // 16 tile columns via shfl_down(width=16), then one atomicMax per segment
  // head. Valid: runs are contiguous, length <= k <= 12 < 16, values >= 0. ----
  {
    const int b = waveBase / nk;                // nk % 16 == 0 -> uniform b
    const int remBase = waveBase - b * nk;
    const int rem = remBase + sN;
    const int n = rem / k;
    const int kk = rem - n * k;
    const bool head = (sN == 0) || (kk == 0);
    bool same[4];
#pragma unroll
    for (int i = 0; i < 4; ++i) {
      const int d = 1 << i;
      same[i] = (sN + d < 16) && ((rem + d) / k == n);
    }
#pragma unroll
    for (int mt = 0; mt < 4; ++mt)
#pragma unroll
      for (int r = 0; r < 8; ++r) {
        const int ch = mt * 16 + r + half * 8;
        float v = acc2[mt][r] * scale2[ch] + shift2[ch];
        v = v > 0.f ? v : 0.f;
#pragma unroll
        for (int i = 0; i < 4; ++i) {
          const float o = __shfl_down(v, 1u << i, 16);
          v = fmaxf(v, same[i] ? o : 0.0f);
        }
        if (head) {
          unsigned int* dst =
              (unsigned int*)&out[((size_t)b * 128 + chOff + ch) * NPTS + n];
          atomicMax(dst, __float_as_uint(v));
        }
      }
  }
}

// ---------------------------------------------------------------------------
extern "C" void kernel_launch(void* const* d_in, const int* in_sizes, int n_in,
                              void* d_out, int out_size, void* d_ws,
                              size_t ws_size, hipStream_t stream) {
  (void)in_sizes; (void)n_in; (void)out_size; (void)ws_size;
  const float* x   = (const float*)d_in[0];
  const float* pos = (const float*)d_in[1];
  const float* Ws1 = (const float*)d_in[2];
  const float* bs1 = (const float*)d_in[3];
  const float* gs1 = (const float*)d_in[4];
  const float* ts1 = (const float*)d_in[5];
  const float* Ws2 = (const float*)d_in[6];
  const float* bs2 = (const float*)d_in[7];
  const float* gs2 = (const float*)d_in[8];
  const float* ts2 = (const float*)d_in[9];
  const float* Wl1 = (const float*)d_in[10];
  const float* bl1 = (const float*)d_in[11];
  const float* gl1 = (const float*)d_in[12];
  const float* tl1 = (const float*)d_in[13];
  const float* Wl2 = (const float*)d_in[14];
  const float* bl2 = (const float*)d_in[15];
  const float* gl2 = (const float*)d_in[16];
  const float* tl2 = (const float*)d_in[17];
  float* out = (float*)d_out;

  int* idxS = (int*)d_ws;
  int* idxL = idxS + (size_t)BATCH * NPTS * KS;
  float* stats = (float*)(idxL + (size_t)BATCH * NPTS * KL);
  // per stream: sum1P[4096] | sq1P[4096] | sum2P[4096] | sq2P[4096] |
  //             scale1[64] | shift1[64] | scale2[64] | shift2[64]
  float* stS = stats;
  float* stL = stats + STREAM_STRIDE;

  zero_kernel<<<(OUTSZ + 255) / 256, 256, 0, stream>>>(out, stats);
  knn_kernel<<<BATCH * NPTS / 256, 256, 0, stream>>>(pos, idxS, idxL);

  const float cntS = (float)((size_t)BATCH * NPTS * KS);
  const float cntL = (float)((size_t)BATCH * NPTS * KL);
  const int gridS = BATCH * NPTS * KS / 64;  // 3072
  const int gridL = BATCH * NPTS * KL / 64;  // 6144

  // ---------------- short-range stream (k = 6), channels [0, 64) ----------
  edgeconv_pass<0><<<gridS, 128, 0, stream>>>(
      x, idxS, KS, Ws1, bs1, nullptr, nullptr, Ws2, bs2, nullptr, nullptr,
      stS + 0, stS + 4096, out, 0);
  bn_finalize_kernel<<<1, 64, 0, stream>>>(stS + 0, stS + 4096, gs1, ts1, cntS,
                                           stS + 16384, stS + 16448);
  edgeconv_pass<1><<<gridS, 128, 0, stream>>>(
      x, idxS, KS, Ws1, bs1, stS + 16384, stS + 16448, Ws2, bs2, nullptr,
      nullptr, stS + 8192, stS + 12288, out, 0);
  bn_finalize_kernel<<<1, 64, 0, stream>>>(stS + 8192, stS + 12288, gs2, ts2,
                                           cntS, stS + 16512, stS + 16576);
  edgeconv_pass<2><<<gridS, 128, 0, stream>>>(
      x, idxS, KS, Ws1, bs1, stS + 16384, stS + 16448, Ws2, bs2, stS + 16512,
      stS + 16576, nullptr, nullptr, out, 0);

  // ---------------- long-range stream (k = 12), channels [64, 128) --------
  edgeconv_pass<0><<<gridL, 128, 0, stream>>>(
      x, idxL, KL, Wl1, bl1, nullptr, nullptr, Wl2, bl2, nullptr, nullptr,
      stL + 0, stL + 4096, out, 64);
  bn_finalize_kernel<<<1, 64, 0, stream>>>(stL + 0, stL + 4096, gl1, tl1, cntL,
                                           stL + 16384, stL + 16448);
  edgeconv_pass<1><<<gridL, 128, 0, stream>>>(
      x, idxL, KL, Wl1, bl1, stL + 16384, stL + 16448, Wl2, bl2, nullptr,
      nullptr, stL + 8192, stL + 12288, out, 64);
  bn_finalize_kernel<<<1, 64, 0, stream>>>(stL + 8192, stL + 12288, gl2, tl2,
                                           cntL, stL + 16512, stL + 16576);
  edgeconv_pass<2><<<gridL, 128, 0, stream>>>(
      x, idxL, KL, Wl1, bl1, stL + 16384, stL + 16448, Wl2, bl2, stL + 16512,
      stL + 16576, nullptr, nullptr, out, 64);
}